// HPWL_33767032881789
// MI455X (gfx1250) — compile-verified
//
#include <hip/hip_runtime.h>

// ---------------------------------------------------------------------------
// HPWL (half-perimeter wirelength) for 1.4M nets / 7M pins.
// Memory-bound gather + segmented min/max + deterministic sum reduction.
// Working set (~90 MB) fits in MI455X's 192 MB L2 -> graph replays run from L2.
//
// Hot path: branchless 8-wide unrolled gather per net. All 8 index loads are
// issued in one clause, then all 16 coordinate gathers in one clause (two
// s_wait_loadcnt rounds total instead of ~deg serialized rounds). Lanes with
// deg < 8 re-gather the last pin via a clamped offset -- idempotent for
// min/max and an L0 hit, so no extra memory traffic.
// ---------------------------------------------------------------------------

#define BLOCK_NETS 256   // 8 wave32 waves per workgroup
#define RED_BLOCK 1024   // 32 waves for the final fold
#define MAXD 8           // degrees in this workload are 2..8; valid ones <= 7

__device__ __forceinline__ float wave32_sum(float v) {
#pragma unroll
    for (int off = 16; off > 0; off >>= 1) {
        v += __shfl_xor(v, off, 32);
    }
    return v;
}

__global__ void __launch_bounds__(BLOCK_NETS)
hpwl_per_net_kernel(const float* __restrict__ pos,
                    const int* __restrict__ flat_netpin,
                    const int* __restrict__ netpin_start,
                    const int* __restrict__ ignore_deg_p,
                    float* __restrict__ block_partials,
                    int num_nets, int num_pins) {
    const int net = blockIdx.x * BLOCK_NETS + threadIdx.x;
    const int ignore_deg = ignore_deg_p[0];

    float h = 0.0f;
    if (net < num_nets) {
        const int s   = netpin_start[net];
        const int e   = netpin_start[net + 1];
        const int deg = e - s;
        if (deg > 0 && deg <= ignore_deg) {
            float xmin =  __builtin_inff();
            float xmax = -__builtin_inff();
            float ymin =  __builtin_inff();
            float ymax = -__builtin_inff();

            if (deg <= MAXD) {
                // ---- branchless batched path (all nets in this workload) ----
                int pin[MAXD];
#pragma unroll
                for (int i = 0; i < MAXD; ++i) {
                    const int off = (i < deg) ? i : (deg - 1);  // clamp: deg >= 1
                    pin[i] = flat_netpin[s + off];
                }
                float xs[MAXD], ys[MAXD];
#pragma unroll
                for (int i = 0; i < MAXD; ++i) {
                    xs[i] = pos[pin[i]];
                    ys[i] = pos[num_pins + pin[i]];
                }
#pragma unroll
                for (int i = 0; i < MAXD; ++i) {
                    xmin = fminf(xmin, xs[i]);
                    xmax = fmaxf(xmax, xs[i]);
                    ymin = fminf(ymin, ys[i]);
                    ymax = fmaxf(ymax, ys[i]);
                }
            } else {
                // ---- generic fallback for large ignore_net_degree ----
                for (int p = s; p < e; ++p) {
                    const int q = flat_netpin[p];
                    const float x = pos[q];
                    const float y = pos[num_pins + q];
                    xmin = fminf(xmin, x);
                    xmax = fmaxf(xmax, x);
                    ymin = fminf(ymin, y);
                    ymax = fmaxf(ymax, y);
                }
            }
            h = (xmax - xmin) + (ymax - ymin);
        }
    }

    // ---- deterministic in-block reduction: wave32 shuffle + LDS ----
    h = wave32_sum(h);

    __shared__ float wsum[BLOCK_NETS / 32];
    const int lane = threadIdx.x & 31;
    const int wave = threadIdx.x >> 5;
    if (lane == 0) wsum[wave] = h;
    __syncthreads();

    if (wave == 0) {
        float v = (lane < (BLOCK_NETS / 32)) ? wsum[lane] : 0.0f;
        v = wave32_sum(v);
        if (lane == 0) block_partials[blockIdx.x] = v;
    }
}

__global__ void __launch_bounds__(RED_BLOCK)
hpwl_final_reduce_kernel(const float* __restrict__ block_partials,
                         int n, float* __restrict__ out) {
    // Fixed-order strided accumulation -> bitwise deterministic result.
    float v = 0.0f;
    for (int i = threadIdx.x; i < n; i += RED_BLOCK) {
        v += block_partials[i];
    }
    v = wave32_sum(v);

    __shared__ float wsum[RED_BLOCK / 32];
    const int lane = threadIdx.x & 31;
    const int wave = threadIdx.x >> 5;
    if (lane == 0) wsum[wave] = v;
    __syncthreads();

    if (wave == 0) {
        float t = (lane < (RED_BLOCK / 32)) ? wsum[lane] : 0.0f;
        t = wave32_sum(t);
        if (lane == 0) out[0] = t;
    }
}

extern "C" void kernel_launch(void* const* d_in, const int* in_sizes, int n_in,
                              void* d_out, int out_size, void* d_ws, size_t ws_size,
                              hipStream_t stream) {
    const float* pos          = (const float*)d_in[0];
    const int*   flat_netpin  = (const int*)d_in[1];
    const int*   netpin_start = (const int*)d_in[2];
    const int*   ignore_deg_p = (const int*)d_in[3];

    const int num_pins = in_sizes[1];
    const int num_nets = in_sizes[2] - 1;

    float* block_partials = (float*)d_ws;   // nblocks floats (~22 KB) << ws_size
    float* out            = (float*)d_out;

    const int nblocks = (num_nets + BLOCK_NETS - 1) / BLOCK_NETS;

    hpwl_per_net_kernel<<<nblocks, BLOCK_NETS, 0, stream>>>(
        pos, flat_netpin, netpin_start, ignore_deg_p,
        block_partials, num_nets, num_pins);

    hpwl_final_reduce_kernel<<<1, RED_BLOCK, 0, stream>>>(
        block_partials, nblocks, out);
}